// Net_32521492365587
// MI455X (gfx1250) — compile-verified
//
#include <hip/hip_runtime.h>
#include <math.h>

typedef __attribute__((ext_vector_type(2))) float v2f;
typedef __attribute__((ext_vector_type(8))) float v8f;

__device__ __forceinline__ float sigm(float v) { return 1.0f / (1.0f + expf(-v)); }

#define W1_LD 164                 // padded LDS row stride: 36*lrow mod 64 distinct -> no bank conflicts

// Single-workgroup fused kernel: whole net in one launch, intermediates in LDS.
// W_hh_opp is never read: the reference calls the opp LSTM with h0 = c0 = 0,
// so the 200x50 recurrent matmul contributes exactly zero.
__global__ __launch_bounds__(256) void net_fused_kernel(
    const float* __restrict__ x,         // (8)
    const float* __restrict__ h0_gen,    // (10,10)
    const float* __restrict__ c0_gen,    // (10,10)
    const float* __restrict__ W_ih_opp,  // (200,8)
    const float* __restrict__ b_ih_opp,  // (200)
    const float* __restrict__ b_hh_opp,  // (200)
    const float* __restrict__ W_ih_gen,  // (10,40,8)
    const float* __restrict__ W_hh_gen,  // (10,40,10)
    const float* __restrict__ b_ih_gen,  // (10,40)
    const float* __restrict__ b_hh_gen,  // (10,40)
    const float* __restrict__ W1,        // (75,150)
    const float* __restrict__ b1,        // (75)
    const float* __restrict__ W2,        // (75)
    const float* __restrict__ b2,        // (1)
    float* __restrict__ out)             // (1)
{
    __shared__ float s_W1[80 * W1_LD];  // W1 zero-padded to 80 x 164 (~52.5 KB)
    __shared__ float s_x[8];
    __shared__ float s_h0[100];
    __shared__ float s_go[200];         // opp gates
    __shared__ float s_gg[400];         // gen gates (10 cells x 40)
    __shared__ float s_feat[160];       // feat padded to 160 (indices 150..159 = 0)
    __shared__ float s_b1[80];          // b1 zero-padded
    __shared__ float s_z[80];           // W1@feat + b1
    __shared__ float s_prod[80];        // W2[j]*z[j]

    const int tid = threadIdx.x;

    if (tid < 8)   s_x[tid]  = x[tid];
    if (tid < 100) s_h0[tid] = h0_gen[tid];
    if (tid < 80)  s_b1[tid] = (tid < 75) ? b1[tid] : 0.0f;

    // Stage W1 into padded LDS (coalesced; zeros in the pad region). No other
    // phase depends on it until the WMMA phase, so it shares the first barrier.
    for (int idx = tid; idx < 80 * W1_LD; idx += 256) {
        const int r = idx / W1_LD, c = idx - r * W1_LD;
        float v = 0.0f;
        if (r < 75 && c < 150) v = W1[r * 150 + c];
        s_W1[idx] = v;
    }
    __syncthreads();

    // ---- Phase 1: all gate pre-activations (600 dot products, len 8 or 18) ----
    for (int k = tid; k < 200; k += 256) {
        float acc = b_ih_opp[k] + b_hh_opp[k];   // Whh@0 == 0
        const float* w = W_ih_opp + k * 8;
        #pragma unroll
        for (int i = 0; i < 8; ++i) acc += w[i] * s_x[i];
        s_go[k] = acc;
    }
    for (int k = tid; k < 400; k += 256) {
        const int g = k / 40;
        float acc = b_ih_gen[k] + b_hh_gen[k];
        const float* wi = W_ih_gen + k * 8;
        #pragma unroll
        for (int i = 0; i < 8; ++i) acc += wi[i] * s_x[i];
        const float* wh = W_hh_gen + k * 10;
        const float* h0 = s_h0 + g * 10;
        #pragma unroll
        for (int i = 0; i < 10; ++i) acc += wh[i] * h0[i];
        s_gg[k] = acc;
    }
    __syncthreads();

    // ---- Phase 2: LSTM cell nonlinearities -> feat (h_gen[100] ++ h_opp[50]) ----
    if (tid < 50) {  // h_opp: c0 = 0 so c = sig(i)*tanh(g)
        float ig = sigm(s_go[tid]);
        float gg = tanhf(s_go[100 + tid]);
        float og = sigm(s_go[150 + tid]);
        s_feat[100 + tid] = og * tanhf(ig * gg);
    }
    if (tid >= 64 && tid < 164) {  // h_gen, gate order i,f,g,o within each cell's 40
        const int idx = tid - 64;          // 0..99
        const int g = idx / 10, j = idx % 10;
        float ig = sigm(s_gg[g * 40 + j]);
        float fg = sigm(s_gg[g * 40 + 10 + j]);
        float gg = tanhf(s_gg[g * 40 + 20 + j]);
        float og = sigm(s_gg[g * 40 + 30 + j]);
        float c  = fg * c0_gen[idx] + ig * gg;
        s_feat[idx] = og * tanhf(c);
    }
    if (tid >= 200 && tid < 210) s_feat[150 + (tid - 200)] = 0.0f;  // K padding
    __syncthreads();

    // ---- Phase 3: z = W1 @ feat + b1 via V_WMMA_F32_16X16X4_F32 ----
    // Waves 0..4 each own a 16-row M-tile; K accumulated in 40 branch-free steps.
    // A(16x4): lane L<16 -> M=L, K=0/1 in v0/v1; lanes 16..31 -> K=2/3.
    // B(4x16): only column N=0 nonzero = feat[k0..k0+3]; same row striping as A.
    const int wave = tid >> 5;
    const int lane = tid & 31;
    if (wave < 5) {
        const int  half  = lane >> 4;               // K-pair select (0 or 1)
        const int  lrow  = lane & 15;               // M within tile / N for B
        const bool isN0  = (lrow == 0);             // lanes carrying B column 0
        const float* aRow = s_W1 + (wave * 16 + lrow) * W1_LD + 2 * half;
        const float* fPtr = s_feat + 2 * half;
        v8f acc = {};
        #pragma unroll
        for (int kb = 0; kb < 40; ++kb) {
            const int k0 = kb * 4;
            v2f A; A.x = aRow[k0]; A.y = aRow[k0 + 1];      // ds_load_b64
            const float f0 = fPtr[k0], f1 = fPtr[k0 + 1];   // ds_load_b64 (broadcast)
            v2f B; B.x = isN0 ? f0 : 0.0f; B.y = isN0 ? f1 : 0.0f;  // v_cndmask
            acc = __builtin_amdgcn_wmma_f32_16x16x4_f32(
                /*neg_a=*/false, A, /*neg_b=*/false, B,
                /*c_mod=*/(short)0, acc, /*reuse_a=*/false, /*reuse_b=*/false);
        }
        // D column N=0 lives in lanes 0 (M=0..7 over acc[0..7]) and 16 (M=8..15).
        if (isN0) {
            #pragma unroll
            for (int r = 0; r < 8; ++r) {
                const int mm = wave * 16 + half * 8 + r;    // < 80, pad-safe
                s_z[mm] = acc[r] + s_b1[mm];
            }
        }
    }
    __syncthreads();

    // ---- Phase 4: out = W2 . z + b2 (serial sum for bit-determinism) ----
    if (tid < 75) s_prod[tid] = W2[tid] * s_z[tid];
    __syncthreads();
    if (tid == 0) {
        float s = b2[0];
        for (int j = 0; j < 75; ++j) s += s_prod[j];
        out[0] = s;
    }
}

extern "C" void kernel_launch(void* const* d_in, const int* in_sizes, int n_in,
                              void* d_out, int out_size, void* d_ws, size_t ws_size,
                              hipStream_t stream) {
    (void)in_sizes; (void)n_in; (void)d_ws; (void)ws_size; (void)out_size;
    const float* x        = (const float*)d_in[0];
    const float* h0_gen   = (const float*)d_in[1];
    const float* c0_gen   = (const float*)d_in[2];
    const float* W_ih_opp = (const float*)d_in[3];
    // d_in[4] = W_hh_opp: provably multiplied by zero state -> never loaded
    const float* b_ih_opp = (const float*)d_in[5];
    const float* b_hh_opp = (const float*)d_in[6];
    const float* W_ih_gen = (const float*)d_in[7];
    const float* W_hh_gen = (const float*)d_in[8];
    const float* b_ih_gen = (const float*)d_in[9];
    const float* b_hh_gen = (const float*)d_in[10];
    const float* W1       = (const float*)d_in[11];
    const float* b1       = (const float*)d_in[12];
    const float* W2       = (const float*)d_in[13];
    const float* b2       = (const float*)d_in[14];
    float* out = (float*)d_out;

    net_fused_kernel<<<1, 256, 0, stream>>>(
        x, h0_gen, c0_gen, W_ih_opp, b_ih_opp, b_hh_opp,
        W_ih_gen, W_hh_gen, b_ih_gen, b_hh_gen, W1, b1, W2, b2, out);
}